// GNN_17935783428252
// MI455X (gfx1250) — compile-verified
//
#include <hip/hip_runtime.h>
#include <stdint.h>

typedef float v2f __attribute__((ext_vector_type(2)));
typedef float v8f __attribute__((ext_vector_type(8)));

// ---------- order-preserving float <-> uint encoding (for max reductions) ----
__device__ __forceinline__ unsigned encf(float f) {
    unsigned u = __float_as_uint(f);
    return (u & 0x80000000u) ? ~u : (u | 0x80000000u);
}
__device__ __forceinline__ float decf(unsigned u) {
    return __uint_as_float((u & 0x80000000u) ? (u & 0x7FFFFFFFu) : ~u);
}

// ---------- init: zero acc buffer, deg=1.0 (self loop), encOut=-inf, minmax --
__global__ void k_init(float* __restrict__ acc, float* __restrict__ deg,
                       unsigned* __restrict__ encOut, unsigned* __restrict__ mm,
                       int n) {
    int t = blockIdx.x * blockDim.x + threadIdx.x;
    if (t < n * 64) acc[t] = 0.0f;
    if (t < 2 * n) deg[t] = 1.0f;               // deg1 then deg2, "+1" folded in
    if (t < 4096) encOut[t] = 0x007FFFFFu;      // enc(-inf)
    if (t == 0) { mm[0] = 0u; mm[1] = 0xFFFFFFFFu; }
}

// ---------- global min/max of edge_attr[:,1] --------------------------------
__global__ __launch_bounds__(256)
void k_minmax(const float* __restrict__ ea, int E, unsigned* __restrict__ mm) {
    __shared__ unsigned smax[256], smin[256];
    unsigned lmax = 0u, lmin = 0xFFFFFFFFu;
    for (int i = blockIdx.x * blockDim.x + threadIdx.x; i < E;
         i += gridDim.x * blockDim.x) {
        unsigned e = encf(ea[2 * i + 1]);
        lmax = lmax > e ? lmax : e;
        lmin = lmin < e ? lmin : e;
    }
    smax[threadIdx.x] = lmax; smin[threadIdx.x] = lmin;
    __syncthreads();
    for (int s = 128; s > 0; s >>= 1) {
        if (threadIdx.x < s) {
            if (smax[threadIdx.x + s] > smax[threadIdx.x]) smax[threadIdx.x] = smax[threadIdx.x + s];
            if (smin[threadIdx.x + s] < smin[threadIdx.x]) smin[threadIdx.x] = smin[threadIdx.x + s];
        }
        __syncthreads();
    }
    if (threadIdx.x == 0) {
        atomicMax(&mm[0], smax[0]);
        atomicMin(&mm[1], smin[0]);
    }
}

// ---------- decode wmax and 1/(wmax-wmin) -----------------------------------
__global__ void k_scale(const unsigned* __restrict__ mm, float* __restrict__ s) {
    float wmax = decf(mm[0]);
    float wmin = decf(mm[1]);
    s[0] = wmax;
    s[1] = 1.0f / (wmax - wmin);
}

// ---------- weighted degree accumulation (both layers in one pass) ----------
__global__ void k_deg(const int* __restrict__ ei, const float* __restrict__ ea,
                      const unsigned char* __restrict__ keep, int E,
                      const float* __restrict__ s,
                      float* __restrict__ deg1, float* __restrict__ deg2) {
    int e = blockIdx.x * blockDim.x + threadIdx.x;
    if (e >= E) return;
    int col = ei[E + e];
    float ew = (s[0] - ea[2 * e + 1]) * s[1];
    float ewd = keep[e] ? ew : 0.0f;
    atomicAdd(&deg1[col], ewd);
    atomicAdd(&deg2[col], ew);
}

// ---------- deg -> rsqrt(deg), in place over both arrays (length 2N) --------
__global__ void k_rsqrt(float* __restrict__ deg, int n2) {
    int t = blockIdx.x * blockDim.x + threadIdx.x;
    if (t < n2) deg[t] = rsqrtf(deg[t]);
}

// ---------- WMMA GEMM: Y[n x 64] = X[n x 64] @ W[64 x 64] -------------------
// Block = 256 threads = 8 waves; block covers 32 rows x 64 cols.
// Each wave computes one 16x16 tile via 16 x V_WMMA_F32_16X16X4_F32 steps.
// A-frag (16x4 f32): lane m(=lane&15), kh(=lane>>4): v0=K(4k+2kh), v1=K(4k+2kh+1)
// B-frag (4x16 f32): same K mapping, N = lane&15
// C/D   (16x16 f32): VGPR r -> M = r + 8*kh, N = lane&15
__global__ __launch_bounds__(256)
void k_gemm(const float* __restrict__ X, const float* __restrict__ W,
            float* __restrict__ Y, int nrows) {
    __shared__ float sX[32][65];
    __shared__ float sW[64][65];
    const int tid  = threadIdx.x;
    const int row0 = blockIdx.x * 32;

    for (int i = tid; i < 4096; i += 256) sW[i >> 6][i & 63] = W[i];
    for (int i = tid; i < 2048; i += 256) {
        int r = i >> 6, c = i & 63;
        int gr = row0 + r;
        sX[r][c] = (gr < nrows) ? X[gr * 64 + c] : 0.0f;
    }
    __syncthreads();

    const int wave = tid >> 5;
    const int lane = tid & 31;
    const int rt = wave >> 2;          // row tile 0..1
    const int ct = wave & 3;           // col tile 0..3
    const int m  = lane & 15;
    const int kh = lane >> 4;          // 0 or 1

    v8f c = {};
    #pragma unroll
    for (int ks = 0; ks < 16; ++ks) {
        const int kb = ks * 4 + kh * 2;
        v2f a, b;
        a.x = sX[rt * 16 + m][kb];
        a.y = sX[rt * 16 + m][kb + 1];
        b.x = sW[kb][ct * 16 + m];
        b.y = sW[kb + 1][ct * 16 + m];
        c = __builtin_amdgcn_wmma_f32_16x16x4_f32(
                false, a, false, b, (short)0, c, false, false);
    }

    const int col = ct * 16 + m;
    #pragma unroll
    for (int r = 0; r < 8; ++r) {
        int grow = row0 + rt * 16 + kh * 8 + r;
        if (grow < nrows) Y[grow * 64 + col] = c[r];
    }
}

// ---------- edge scatter: acc[col] += xw[row] * dinv[row]*ew*dinv[col] ------
// One thread per (edge, feature): feature = tid&63 -> coalesced 128B per wave.
__global__ __launch_bounds__(256)
void k_scatter(const float* __restrict__ xw, float* __restrict__ acc,
               const int* __restrict__ ei, const float* __restrict__ ea,
               const unsigned char* __restrict__ keep, int use_drop,
               const float* __restrict__ s, const float* __restrict__ dinv,
               int E) {
    unsigned t = blockIdx.x * blockDim.x + threadIdx.x;
    int e = (int)(t >> 6);
    int d = (int)(t & 63u);
    if (e >= E) return;
    if (use_drop && !keep[e]) return;            // ew_drop == 0 -> no contribution
    float ew = (s[0] - ea[2 * e + 1]) * s[1];
    int row = ei[e];
    int col = ei[E + e];
    float norm = dinv[row] * ew * dinv[col];
    atomicAdd(&acc[col * 64 + d], xw[row * 64 + d] * norm);
}

// ---------- finalize layer 1: x1 = relu(acc + xw*dinv^2 + b1), in place -----
__global__ void k_final1(float* __restrict__ acc, const float* __restrict__ xw,
                         const float* __restrict__ dinv,
                         const float* __restrict__ b1, int n) {
    int t = blockIdx.x * blockDim.x + threadIdx.x;
    if (t >= n * 64) return;
    int i = t >> 6, d = t & 63;
    float di = dinv[i];
    float v = acc[t] + xw[t] * di * di + b1[d];
    acc[t] = v > 0.0f ? v : 0.0f;
}

// ---------- zero acc buffer for layer 2 -------------------------------------
__global__ void k_zero(float* __restrict__ p, int count) {
    int t = blockIdx.x * blockDim.x + threadIdx.x;
    if (t < count) p[t] = 0.0f;
}

// ---------- finalize layer 2 + residual + fused encoded segment_max ---------
__global__ void k_final2(const float* __restrict__ acc, const float* __restrict__ xw,
                         const float* __restrict__ x, const float* __restrict__ dinv,
                         const float* __restrict__ b2, const int* __restrict__ batch,
                         unsigned* __restrict__ encOut, int n) {
    int t = blockIdx.x * blockDim.x + threadIdx.x;
    if (t >= n * 64) return;
    int i = t >> 6, d = t & 63;
    float di = dinv[i];
    float v = acc[t] + xw[t] * di * di + b2[d] + x[t];
    atomicMax(&encOut[batch[i] * 64 + d], encf(v));
}

// ---------- decode segment_max result to d_out ------------------------------
__global__ void k_decode(const unsigned* __restrict__ encOut, float* __restrict__ out,
                         int count) {
    int t = blockIdx.x * blockDim.x + threadIdx.x;
    if (t < count) out[t] = decf(encOut[t]);
}

extern "C" void kernel_launch(void* const* d_in, const int* in_sizes, int n_in,
                              void* d_out, int out_size, void* d_ws, size_t ws_size,
                              hipStream_t stream) {
    // inputs: x, edge_index, edge_attr, batch, keep_mask, W1, b1, W2, b2
    const float* x          = (const float*)d_in[0];
    const int*   edge_index = (const int*)d_in[1];
    const float* edge_attr  = (const float*)d_in[2];
    const int*   batch      = (const int*)d_in[3];
    const unsigned char* keep = (const unsigned char*)d_in[4];
    const float* W1 = (const float*)d_in[5];
    const float* b1 = (const float*)d_in[6];
    const float* W2 = (const float*)d_in[7];
    const float* b2 = (const float*)d_in[8];
    float* out = (float*)d_out;

    const int N = in_sizes[0] / 64;       // 75000
    const int E = in_sizes[2] / 2;        // 1200000
    const int ND = N * 64;

    // workspace partition (256B aligned)
    size_t off = 0;
    auto carve = [&](size_t bytes) -> char* {
        off = (off + 255) & ~(size_t)255;
        char* p = (char*)d_ws + off;
        off += bytes;
        return p;
    };
    float*    bufA   = (float*)carve((size_t)ND * 4);   // xw1 / xw2
    float*    bufB   = (float*)carve((size_t)ND * 4);   // acc1 / x1 / acc2
    float*    deg    = (float*)carve((size_t)2 * N * 4); // deg1 | deg2 -> dinv
    float*    deg1   = deg;
    float*    deg2   = deg + N;
    unsigned* mm     = (unsigned*)carve(2 * 4);
    float*    sca    = (float*)carve(2 * 4);
    unsigned* encOut = (unsigned*)carve(4096 * 4);

    const int TB = 256;
    const int gElem  = (ND + TB - 1) / TB;              // N*64 threads
    const int gEdge  = (E + TB - 1) / TB;               // E threads
    const int gEdgeF = (int)(((long long)E * 64 + TB - 1) / TB); // E*64 threads
    const int gGemm  = (N + 31) / 32;

    // 0) init
    k_init<<<gElem, TB, 0, stream>>>(bufB, deg, encOut, mm, N);
    // 1) min/max of raw weights
    k_minmax<<<1024, TB, 0, stream>>>(edge_attr, E, mm);
    // 2) scale scalars
    k_scale<<<1, 1, 0, stream>>>(mm, sca);
    // 3) weighted degrees (both layers)
    k_deg<<<gEdge, TB, 0, stream>>>(edge_index, edge_attr, keep, E, sca, deg1, deg2);
    // 4) dinv = rsqrt(deg)
    k_rsqrt<<<(2 * N + TB - 1) / TB, TB, 0, stream>>>(deg, 2 * N);
    // 5) xw1 = x @ W1  (WMMA)
    k_gemm<<<gGemm, TB, 0, stream>>>(x, W1, bufA, N);
    // 6) scatter layer 1 (with dropout mask)
    k_scatter<<<gEdgeF, TB, 0, stream>>>(bufA, bufB, edge_index, edge_attr,
                                         keep, 1, sca, deg1, E);
    // 7) x1 = relu(acc1 + xw1*dinv1^2 + b1)   (in place in bufB)
    k_final1<<<gElem, TB, 0, stream>>>(bufB, bufA, deg1, b1, N);
    // 8) xw2 = x1 @ W2  (WMMA)
    k_gemm<<<gGemm, TB, 0, stream>>>(bufB, W2, bufA, N);
    // 9) acc2 = 0
    k_zero<<<gElem, TB, 0, stream>>>(bufB, ND);
    // 10) scatter layer 2 (no dropout)
    k_scatter<<<gEdgeF, TB, 0, stream>>>(bufA, bufB, edge_index, edge_attr,
                                         keep, 0, sca, deg2, E);
    // 11) x2 = acc2 + xw2*dinv2^2 + b2 + x ; fused encoded segment_max
    k_final2<<<gElem, TB, 0, stream>>>(bufB, bufA, x, deg2, b2, batch, encOut, N);
    // 12) decode to output (G*D = 4096 floats)
    k_decode<<<(out_size + TB - 1) / TB, TB, 0, stream>>>(encOut, out, out_size);
}